// OptimizedE8Layer_40269613367378
// MI455X (gfx1250) — compile-verified
//
#include <hip/hip_runtime.h>
#include <hip/hip_bf16.h>
#include <math.h>

typedef float v2f __attribute__((ext_vector_type(2)));
typedef float v8f __attribute__((ext_vector_type(8)));

#define DFEAT 128

// ---------------- zero scratch ----------------
__global__ void gcn_zero_f32(float* __restrict__ p, size_t n) {
    size_t i = (size_t)blockIdx.x * blockDim.x + threadIdx.x;
    size_t stride = (size_t)gridDim.x * blockDim.x;
    for (; i < n; i += stride) p[i] = 0.0f;
}

// ---------------- degree count over col ----------------
__global__ void gcn_degree(const int* __restrict__ col, float* __restrict__ deg, int nEdges) {
    int e = blockIdx.x * blockDim.x + threadIdx.x;
    if (e < nEdges) atomicAdd(&deg[col[e]], 1.0f);
}

// ---------------- deg_inv_sqrt ----------------
__global__ void gcn_dis(const float* __restrict__ deg, float* __restrict__ dis, int n) {
    int i = blockIdx.x * blockDim.x + threadIdx.x;
    if (i < n) {
        float d = deg[i];
        dis[i] = (d > 0.0f) ? rsqrtf(d) : 0.0f;  // deg is a count: max(deg,1)==deg when >0
    }
}

// ---------------- SpMM scatter: one wave32 per edge, float4 per lane ----------------
__global__ void gcn_scatter(const float* __restrict__ x,
                            const int* __restrict__ row, const int* __restrict__ col,
                            const float* __restrict__ dis,
                            float* __restrict__ agg, int nEdges) {
    int gid = blockIdx.x * blockDim.x + threadIdx.x;
    int e = gid >> 5;
    int lane = threadIdx.x & 31;
    if (e >= nEdges) return;
    int r = row[e];
    int c = col[e];
    float val = dis[r] * dis[c];
    const float4* xs = (const float4*)(x + (size_t)c * DFEAT);
    float4 v = xs[lane];                       // 32 lanes x 16B = 512B row, coalesced
    float* dst = agg + (size_t)r * DFEAT + lane * 4;
    atomicAdd(dst + 0, val * v.x);
    atomicAdd(dst + 1, val * v.y);
    atomicAdd(dst + 2, val * v.z);
    atomicAdd(dst + 3, val * v.w);
}

// ---------------- dense GEMM: out[N,128] = agg[N,128] @ W[128,128]^T ----------------
// One wave per 16-row tile; 8 N-tiles of 16 cover D_OUT=128.
// V_WMMA_F32_16X16X4_F32:  D(16x16) = A(16x4) x B(4x16) + C
// fp32 A layout: lanes 0-15 M=0..15 {K=k0,k0+1}; lanes 16-31 M=0..15 {K=k0+2,k0+3}
// fp32 C/D layout: VGPR v -> row v + 8*(lane>=16), col = lane%16
__global__ __launch_bounds__(128) void gcn_gemm_wmma(const float* __restrict__ agg,
                                                     const float* __restrict__ W,
                                                     float* __restrict__ out, int nTiles) {
    int wave = (int)((blockIdx.x * blockDim.x + threadIdx.x) >> 5);
    int lane = threadIdx.x & 31;
    if (wave >= nTiles) return;            // uniform per-wave: EXEC stays all-1s for WMMA

    const int half = lane >> 4;            // 0 or 1
    const int lm   = lane & 15;
    const int mBase = wave * 16;

    const float* aRow = agg + (size_t)(mBase + lm) * DFEAT + 2 * half;

    v8f acc[8] = {};                       // 8 x 16x16 f32 accumulators = full 16x128 tile

    for (int k0 = 0; k0 < DFEAT; k0 += 4) {
        v2f a;
        a.x = aRow[k0];
        a.y = aRow[k0 + 1];
#pragma unroll
        for (int n = 0; n < 8; ++n) {
            const float* bPtr = W + (size_t)(n * 16 + lm) * DFEAT + k0 + 2 * half;
            v2f b;
            b.x = bPtr[0];                 // B[k][n] = W[n][k], contiguous in k
            b.y = bPtr[1];
            acc[n] = __builtin_amdgcn_wmma_f32_16x16x4_f32(
                /*neg_a=*/false, a, /*neg_b=*/false, b,
                /*c_mod=*/(short)0, acc[n], /*reuse_a=*/false, /*reuse_b=*/false);
        }
    }

#pragma unroll
    for (int n = 0; n < 8; ++n) {
#pragma unroll
        for (int v = 0; v < 8; ++v) {
            int r = mBase + v + 8 * half;
            out[(size_t)r * DFEAT + n * 16 + lm] = acc[n][v];
        }
    }
}

extern "C" void kernel_launch(void* const* d_in, const int* in_sizes, int n_in,
                              void* d_out, int out_size, void* d_ws, size_t ws_size,
                              hipStream_t stream) {
    (void)n_in; (void)out_size; (void)ws_size;

    const float* x  = (const float*)d_in[0];   // [N,128]
    const float* W  = (const float*)d_in[1];   // [128,128]
    const int*   ei = (const int*)d_in[2];     // [2,E] flat: row then col
    float* out = (float*)d_out;

    const int N = in_sizes[0] / DFEAT;         // 100000
    const int E = in_sizes[2] / 2;             // 1600000

    // scratch layout: agg[N*128] | deg[N] | dis[N]
    float* agg = (float*)d_ws;
    float* deg = agg + (size_t)N * DFEAT;
    float* dis = deg + N;

    const int* rowIdx = ei;
    const int* colIdx = ei + E;

    // 1) zero agg + deg (contiguous region)
    size_t zn = (size_t)N * DFEAT + (size_t)N;
    gcn_zero_f32<<<2048, 256, 0, stream>>>(agg, zn);

    // 2) degree over col
    gcn_degree<<<(E + 255) / 256, 256, 0, stream>>>(colIdx, deg, E);

    // 3) deg_inv_sqrt
    gcn_dis<<<(N + 255) / 256, 256, 0, stream>>>(deg, dis, N);

    // 4) edge scatter: one wave per edge
    long long scatterThreads = (long long)E * 32;
    int scatterBlocks = (int)((scatterThreads + 255) / 256);
    gcn_scatter<<<scatterBlocks, 256, 0, stream>>>(x, rowIdx, colIdx, dis, agg, E);

    // 5) WMMA GEMM: one wave per 16-row tile (N divisible by 16)
    int nTiles = N / 16;                       // 6250
    int waves_per_block = 128 / 32;
    int gemmBlocks = (nTiles + waves_per_block - 1) / waves_per_block;
    gcn_gemm_wmma<<<gemmBlocks, 128, 0, stream>>>(agg, W, out, nTiles);
}